// SoftPeakAwareLoss_71545565217536
// MI455X (gfx1250) — compile-verified
//
#include <hip/hip_runtime.h>

// SoftPeakAwareLoss fused kernel for gfx1250 (MI455X).
// Memory-bound: 283 MB read @ 23.3 TB/s => ~12 us floor. Single fused pass,
// staged through LDS via CDNA5 async global->LDS copies (ASYNCcnt).

#define HOURS 24
#define WIN 2
#define BLOCK_THREADS 128           // 4 waves of 32
#define WAVES_PER_BLOCK 4
#define CHUNKS_PER_WAVE 32
#define TILE_FLOATS (CHUNKS_PER_WAVE * HOURS)   // 768 floats = 3 KB per array

// Get the workgroup-relative LDS byte offset of a pointer derived from __shared__.
__device__ __forceinline__ unsigned lds_off_of(void* p) {
  return (unsigned)(unsigned long long)(__attribute__((address_space(3))) char*)p;
}

// CDNA5 async copy: 16 bytes global -> LDS, tracked by ASYNCcnt.
__device__ __forceinline__ void async_ld16(unsigned lds_off, const void* gaddr) {
  asm volatile("global_load_async_to_lds_b128 %0, %1, off"
               :: "v"(lds_off), "v"(gaddr)
               : "memory");
}

__device__ __forceinline__ void wait_async0() {
  asm volatile("s_wait_asynccnt 0" ::: "memory");
}

__global__ __launch_bounds__(BLOCK_THREADS, 1)
void splk_main(const float* __restrict__ y_pred,
               const float* __restrict__ y_true,
               const float* __restrict__ dmask,
               float* __restrict__ part) {
  __shared__ float smem[WAVES_PER_BLOCK * 3 * TILE_FLOATS];   // 36 KB
  __shared__ float wred[WAVES_PER_BLOCK][4];

  const int tid  = threadIdx.x;
  const int wave = tid >> 5;
  const int lane = tid & 31;

  const long long chunk0 = (long long)blockIdx.x * (BLOCK_THREADS) + (long long)wave * CHUNKS_PER_WAVE;
  const long long g0 = chunk0 * HOURS;   // first float index of this wave's tile

  float* sT = &smem[wave * (3 * TILE_FLOATS)];
  float* sP = sT + TILE_FLOATS;
  float* sD = sT + 2 * TILE_FLOATS;

  // ---- Stage tiles: 6x b128 per array per lane, fully coalesced ----
  {
    const char* gT = (const char*)(y_true + g0) + lane * 16;
    const char* gP = (const char*)(y_pred + g0) + lane * 16;
    const char* gD = (const char*)(dmask  + g0) + lane * 16;
    unsigned oT = lds_off_of(sT) + (unsigned)lane * 16u;
    unsigned oP = lds_off_of(sP) + (unsigned)lane * 16u;
    unsigned oD = lds_off_of(sD) + (unsigned)lane * 16u;
#pragma unroll
    for (int j = 0; j < 6; ++j) {
      async_ld16(oT + j * 512u, gT + j * 512);
      async_ld16(oP + j * 512u, gP + j * 512);
      async_ld16(oD + j * 512u, gD + j * 512);
    }
  }
  wait_async0();   // wave's own async loads complete; only own region is read

  const float* ct = sT + lane * HOURS;
  const float* cp = sP + lane * HOURS;
  const float* cd = sD + lane * HOURS;

  // ---- Pull chunk into registers (16B-aligned b128 LDS reads) ----
  float yt[HOURS], yp[HOURS];
  unsigned maskbits = 0;
#pragma unroll
  for (int k = 0; k < 6; ++k) {
    float4 t4 = ((const float4*)ct)[k];
    float4 p4 = ((const float4*)cp)[k];
    float4 d4 = ((const float4*)cd)[k];
    yt[4*k+0] = t4.x; yt[4*k+1] = t4.y; yt[4*k+2] = t4.z; yt[4*k+3] = t4.w;
    yp[4*k+0] = p4.x; yp[4*k+1] = p4.y; yp[4*k+2] = p4.z; yp[4*k+3] = p4.w;
    maskbits |= (d4.x > 0.5f ? 1u : 0u) << (4*k+0);
    maskbits |= (d4.y > 0.5f ? 1u : 0u) << (4*k+1);
    maskbits |= (d4.z > 0.5f ? 1u : 0u) << (4*k+2);
    maskbits |= (d4.w > 0.5f ? 1u : 0u) << (4*k+3);
  }

  // ---- Pass A: overall MSE partial, maxes, first-argmax of masked true ----
  float s_over = 0.f;
  float mxP = -3.402823466e38f;
  float bm  = -3.402823466e38f;
  int pk = 0;
#pragma unroll
  for (int i = 0; i < HOURS; ++i) {
    float d = yp[i] - yt[i];
    s_over = fmaf(d, d, s_over);
    bool c = (maskbits >> i) & 1u;
    float mt = c ? yt[i] : -1e9f;
    float mp = c ? yp[i] : -1e9f;
    if (mt > bm) { bm = mt; pk = i; }   // strict > keeps first occurrence
    mxP = fmaxf(mxP, mp);
  }
  const float mxT = bm;

  // ---- Pass B: both softmaxes, fused weighted sums ----
  float swT = 0.f, syT = 0.f, stT = 0.f;
  float swP = 0.f, syP = 0.f, stP = 0.f;
#pragma unroll
  for (int i = 0; i < HOURS; ++i) {
    bool c = (maskbits >> i) & 1u;
    float mt = c ? yt[i] : -1e9f;
    float mp = c ? yp[i] : -1e9f;
    float wT = __expf(10.0f * (mt - mxT));   // masked-out -> exp(-1e10) = 0
    float wP = __expf(10.0f * (mp - mxP));
    swT += wT; syT = fmaf(wT, yt[i], syT); stT = fmaf(wT, (float)i, stT);
    swP += wP; syP = fmaf(wP, yp[i], syP); stP = fmaf(wP, (float)i, stP);
  }
  float tv = syT / swT, tt = stT / swT;
  float pv = syP / swP, pt = stP / swP;
  float dmag = pv - tv; float s_mag  = dmag * dmag;
  float dtim = pt - tt; float s_time = dtim * dtim;

  // ---- Pass C: +-WIN window around hard peak (dynamic index -> LDS reads) ----
  float twv[2 * WIN + 1], pwv[2 * WIN + 1];
  bool  vv[2 * WIN + 1];
  float tmax = -3.402823466e38f, pmax = -3.402823466e38f, cnt = 0.f;
#pragma unroll
  for (int o = 0; o < 2 * WIN + 1; ++o) {
    int idx = pk + o - WIN;
    bool v = (idx >= 0) && (idx < HOURS);
    int ic = idx < 0 ? 0 : (idx > HOURS - 1 ? HOURS - 1 : idx);
    float tw = ct[ic];
    float pw = cp[ic];
    twv[o] = tw; pwv[o] = pw; vv[o] = v;
    if (v) { tmax = fmaxf(tmax, tw); pmax = fmaxf(pmax, pw); cnt += 1.f; }
  }
  float denT = tmax + 1e-6f;
  float denP = pmax + 1e-6f;
  float per = 0.f;
#pragma unroll
  for (int o = 0; o < 2 * WIN + 1; ++o) {
    if (vv[o]) {
      float dn = pwv[o] / denP - twv[o] / denT;
      per = fmaf(dn, dn, per);
    }
  }
  per /= cnt;

  // ---- Deterministic block reduction of the 4 partial sums ----
  float a0 = s_over, a1 = s_mag, a2 = s_time, a3 = per;
#pragma unroll
  for (int off = 16; off > 0; off >>= 1) {
    a0 += __shfl_down(a0, off, 32);
    a1 += __shfl_down(a1, off, 32);
    a2 += __shfl_down(a2, off, 32);
    a3 += __shfl_down(a3, off, 32);
  }
  if (lane == 0) { wred[wave][0] = a0; wred[wave][1] = a1; wred[wave][2] = a2; wred[wave][3] = a3; }
  __syncthreads();
  if (tid == 0) {
    float b0 = 0.f, b1 = 0.f, b2 = 0.f, b3 = 0.f;
#pragma unroll
    for (int w = 0; w < WAVES_PER_BLOCK; ++w) {
      b0 += wred[w][0]; b1 += wred[w][1]; b2 += wred[w][2]; b3 += wred[w][3];
    }
    float* o = part + (size_t)blockIdx.x * 4;
    o[0] = b0; o[1] = b1; o[2] = b2; o[3] = b3;
  }
}

__global__ __launch_bounds__(256, 1)
void splk_final(const float* __restrict__ part, int nblocks,
                long long nBS, long long nBD, float* __restrict__ out) {
  __shared__ float wred[8][4];
  const int tid = threadIdx.x;
  const int wave = tid >> 5;
  const int lane = tid & 31;
  float a0 = 0.f, a1 = 0.f, a2 = 0.f, a3 = 0.f;
  for (int i = tid; i < nblocks; i += 256) {
    const float* p = part + (size_t)i * 4;
    a0 += p[0]; a1 += p[1]; a2 += p[2]; a3 += p[3];
  }
#pragma unroll
  for (int off = 16; off > 0; off >>= 1) {
    a0 += __shfl_down(a0, off, 32);
    a1 += __shfl_down(a1, off, 32);
    a2 += __shfl_down(a2, off, 32);
    a3 += __shfl_down(a3, off, 32);
  }
  if (lane == 0) { wred[wave][0] = a0; wred[wave][1] = a1; wred[wave][2] = a2; wred[wave][3] = a3; }
  __syncthreads();
  if (tid == 0) {
    float b0 = 0.f, b1 = 0.f, b2 = 0.f, b3 = 0.f;
#pragma unroll
    for (int w = 0; w < 8; ++w) {
      b0 += wred[w][0]; b1 += wred[w][1]; b2 += wred[w][2]; b3 += wred[w][3];
    }
    float invBS = 1.0f / (float)nBS;
    float invBD = 1.0f / (float)nBD;
    float loss = 1.0f * b0 * invBS            // ALPHA * L_overall
               + 2.0f * b1 * invBD            // BETA  * L_mag
               + 1.0f * b2 * invBD            // GAMMA * L_time
               + 0.5f * (b3 / ((float)nBD + 1e-6f)); // DELTA * L_shape
    out[0] = loss;
  }
}

extern "C" void kernel_launch(void* const* d_in, const int* in_sizes, int n_in,
                              void* d_out, int out_size, void* d_ws, size_t ws_size,
                              hipStream_t stream) {
  const float* y_pred = (const float*)d_in[0];
  const float* y_true = (const float*)d_in[1];
  const float* dmask  = (const float*)d_in[2];
  float* out  = (float*)d_out;
  float* part = (float*)d_ws;

  const long long n      = (long long)in_sizes[0];  // B*S = 23,592,960
  const long long chunks = n / HOURS;               // B*D = 983,040
  const int nblocks = (int)(chunks / BLOCK_THREADS);// 7680, exact

  splk_main<<<nblocks, BLOCK_THREADS, 0, stream>>>(y_pred, y_true, dmask, part);
  splk_final<<<1, 256, 0, stream>>>(part, nblocks, n, chunks, out);
}